// RPN_52390011076626
// MI455X (gfx1250) — compile-verified
//
#include <hip/hip_runtime.h>
#include <cstdint>
#include <math.h>

#define BLOCK 256
#define NP3   49152              // 128*128*3
#define NP4   12288              // 64*64*3
#define NP5   3072               // 32*32*3
#define NTOT  (NP3 + NP4 + NP5)  // 64512 = 252*256 exactly
#define BATCH 16
#define MGT   100
#define IOU_LOW  0.3f
#define IOU_HIGH 0.6f
#define NEG_BIG  (-1e8f)
#define SCALE_CLAMP 3.3322045f   // log(224/8)

__device__ __forceinline__ void async_g2l_b32(const float* smem_dst, const float* gsrc) {
    uint32_t lds_off = (uint32_t)(uintptr_t)smem_dst;   // AS3 offset (low 32 bits of generic)
    uint64_t gaddr   = (uint64_t)(uintptr_t)gsrc;
    asm volatile("global_load_async_to_lds_b32 %0, %1, off"
                 :
                 : "v"(lds_off), "v"(gaddr)
                 : "memory");
}

__global__ __launch_bounds__(BLOCK)
void rpn_match_kernel(const float* __restrict__ locs_p3,
                      const float* __restrict__ locs_p4,
                      const float* __restrict__ locs_p5,
                      const float* __restrict__ deltas,
                      const float* __restrict__ gt,
                      float* __restrict__ out_prop,
                      float* __restrict__ out_match,
                      float* __restrict__ out_gtd)
{
    // GT rows padded to 8 floats: x1,y1,x2,y2,cls,area,validFlag,pad
    __shared__ __align__(16) float sgt[MGT * 8];

    const int tid = threadIdx.x;
    const int b   = blockIdx.y;
    const int n   = blockIdx.x * BLOCK + tid;   // grid is exact, no bounds check

    // ---- Issue per-thread global loads FIRST so HBM latency overlaps staging ----
    const float* locs;
    int base;
    float area;
    if (n < NP3)            { locs = locs_p3; base = n;             area = 1024.0f;  }
    else if (n < NP3 + NP4) { locs = locs_p4; base = n - NP3;       area = 4096.0f;  }
    else                    { locs = locs_p5; base = n - NP3 - NP4; area = 16384.0f; }
    const int loc = base / 3;
    const int r   = base - loc * 3;             // aspect-ratio index 0..2

    const float2 xy = ((const float2*)locs)[loc];                       // global_load_b64
    const float4 d4 = ((const float4*)deltas)[(size_t)b * NTOT + n];    // global_load_b128

    // ---- Stage this batch's 100x5 GT floats into LDS via async global->LDS ----
    const float* gtb = gt + (size_t)b * (MGT * 5);
    {
        int t = tid;                            // 0..255, always < 500
        int j = t / 5, c = t - j * 5;
        async_g2l_b32(&sgt[j * 8 + c], gtb + t);
        t = tid + BLOCK;                        // 256..511
        if (t < MGT * 5) {
            j = t / 5; c = t - j * 5;
            async_g2l_b32(&sgt[j * 8 + c], gtb + t);
        }
    }
    asm volatile("s_wait_asynccnt 0" ::: "memory");
    __syncthreads();

    // Precompute per-GT area and validity once per block (slots 5,6 -> one b64 read later).
    for (int j = tid; j < MGT; j += BLOCK) {
        float x1 = sgt[j * 8 + 0], y1 = sgt[j * 8 + 1];
        float x2 = sgt[j * 8 + 2], y2 = sgt[j * 8 + 3];
        float cls = sgt[j * 8 + 4];
        sgt[j * 8 + 5] = (x2 - x1) * (y2 - y1);
        sgt[j * 8 + 6] = (cls != -1.0f) ? 1.0f : 0.0f;
    }
    __syncthreads();

    // ---- Anchor geometry ----
    const float ar = 0.5f * (float)(1 << r);    // 0.5, 1.0, 2.0
    const float w  = sqrtf(area / ar);
    const float h  = area / w;
    const float x  = xy.x, y = xy.y;
    const float ax1 = x - 0.5f * w, ay1 = y - 0.5f * h;
    const float ax2 = x + 0.5f * w, ay2 = y + 0.5f * h;
    const float pw  = ax2 - ax1,    ph  = ay2 - ay1;
    const float px  = ax1 + 0.5f * pw;
    const float py  = ay1 + 0.5f * ph;
    const float a1  = pw * ph;

    // ---- apply_deltas (proposal decode) ----
    {
        const float dw = fminf(d4.z, SCALE_CLAMP);
        const float dh = fminf(d4.w, SCALE_CLAMP);
        const float bx = px + pw * d4.x;
        const float by = py + ph * d4.y;
        const float bw = pw * expf(dw);
        const float bh = ph * expf(dh);
        float4 prop;
        prop.x = bx - 0.5f * bw;
        prop.y = by - 0.5f * bh;
        prop.z = bx + 0.5f * bw;
        prop.w = by + 0.5f * bh;
        ((float4*)out_prop)[(size_t)b * NTOT + n] = prop;
    }

    // ---- IoU argmax over 100 GT boxes ----
    // Division-free rational compare; two independent tournament chains (even/odd j)
    // to halve the loop-carried mul->cmp->select dependency chain.
    float bn0 = -1.0f, bd0 = 1.0f;  int bi0 = 0;
    float bn1 = -1.0f, bd1 = 1.0f;  int bi1 = 1;
    #pragma unroll 4
    for (int j = 0; j < MGT; j += 2) {
        {   // even chain
            const float4 g  = *(const float4*)&sgt[j * 8];
            const float2 av = *(const float2*)&sgt[j * 8 + 5];   // {area, valid}
            float ix = fminf(ax2, g.z) - fmaxf(ax1, g.x);
            float iy = fminf(ay2, g.w) - fmaxf(ay1, g.y);
            ix = fmaxf(ix, 0.0f); iy = fmaxf(iy, 0.0f);
            const float inter = ix * iy;
            const float uni   = a1 + av.x - inter;
            const float num = (av.y != 0.0f) ? inter : -1.0f;
            const float den = (av.y != 0.0f) ? uni   :  1.0f;
            if (num * bd0 > bn0 * den) { bn0 = num; bd0 = den; bi0 = j; }
        }
        {   // odd chain
            const float4 g  = *(const float4*)&sgt[(j + 1) * 8];
            const float2 av = *(const float2*)&sgt[(j + 1) * 8 + 5];
            float ix = fminf(ax2, g.z) - fmaxf(ax1, g.x);
            float iy = fminf(ay2, g.w) - fmaxf(ay1, g.y);
            ix = fmaxf(ix, 0.0f); iy = fmaxf(iy, 0.0f);
            const float inter = ix * iy;
            const float uni   = a1 + av.x - inter;
            const float num = (av.y != 0.0f) ? inter : -1.0f;
            const float den = (av.y != 0.0f) ? uni   :  1.0f;
            if (num * bd1 > bn1 * den) { bn1 = num; bd1 = den; bi1 = j + 1; }
        }
    }
    // Merge chains; on exact rational tie prefer the lower index (jnp.argmax semantics).
    const float c10 = bn1 * bd0;          // chain1 value cross-multiplied
    const float c01 = bn0 * bd1;          // chain0 value cross-multiplied
    const bool take1 = (c10 > c01) || ((c10 == c01) && (bi1 < bi0));
    const float bn = take1 ? bn1 : bn0;
    const float bd = take1 ? bd1 : bd0;
    const int bidx = take1 ? bi1 : bi0;
    const float q  = bn / bd;             // same rounding as reference for the winner

    // ---- matched GT row + background/neutral overrides ----
    const float4 gb = *(const float4*)&sgt[bidx * 8];
    float m0 = gb.x, m1 = gb.y, m2 = gb.z, m3 = gb.w;
    float m4 = sgt[bidx * 8 + 4];
    if (q <= IOU_LOW) {
        m0 = m1 = m2 = m3 = m4 = -1.0f;          // background
    } else if (q < IOU_HIGH) {
        m0 = m1 = m2 = m3 = m4 = NEG_BIG;        // neutral
    }
    {
        const size_t mo = ((size_t)b * NTOT + n) * 5;
        out_match[mo + 0] = m0;
        out_match[mo + 1] = m1;
        out_match[mo + 2] = m2;
        out_match[mo + 3] = m3;
        out_match[mo + 4] = m4;
    }

    // ---- get_deltas (regression targets from matched row) ----
    float4 gd;
    if (m0 == -1.0f) {
        gd.x = gd.y = gd.z = gd.w = NEG_BIG;
    } else {
        const float gw = m2 - m0;
        const float gh = m3 - m1;
        const float gx = m0 + 0.5f * gw;
        const float gy = m1 + 0.5f * gh;
        gd.x = (gx - px) / pw;
        gd.y = (gy - py) / ph;
        gd.z = logf(gw / pw);    // neutral rows: log(0/pw) -> -inf, same as JAX
        gd.w = logf(gh / ph);
    }
    ((float4*)out_gtd)[(size_t)b * NTOT + n] = gd;
}

extern "C" void kernel_launch(void* const* d_in, const int* in_sizes, int n_in,
                              void* d_out, int out_size, void* d_ws, size_t ws_size,
                              hipStream_t stream) {
    const float* locs_p3 = (const float*)d_in[0];
    const float* locs_p4 = (const float*)d_in[1];
    const float* locs_p5 = (const float*)d_in[2];
    const float* deltas  = (const float*)d_in[3];
    const float* gt      = (const float*)d_in[4];

    float* out       = (float*)d_out;
    float* out_prop  = out;                                   // (B,N,4)
    float* out_match = out_prop  + (size_t)BATCH * NTOT * 4;  // (B,N,5)
    float* out_gtd   = out_match + (size_t)BATCH * NTOT * 5;  // (B,N,4)

    dim3 grid(NTOT / BLOCK, BATCH);   // 252 x 16 full blocks
    rpn_match_kernel<<<grid, BLOCK, 0, stream>>>(
        locs_p3, locs_p4, locs_p5, deltas, gt, out_prop, out_match, out_gtd);
}